// SplitHead_62517543960955
// MI455X (gfx1250) — compile-verified
//
#include <hip/hip_runtime.h>

// Problem constants (from the reference)
#define B_  8
#define C_  64
#define H_  200
#define W_  320
#define HW_ 64000          // H_*W_
#define N_  48
#define WD_ 80             // W_/4

// GEMM tiling for the big kernel
#define TILE_COLS       128
#define TILES_PER_BLOCK 10
#define LDSROW          136  // floats per channel row (128 cols + 8 pad)
                             // 2*136 mod 64 == 16 -> lane halves hit disjoint banks

typedef float v2f __attribute__((ext_vector_type(2)));
typedef float v8f __attribute__((ext_vector_type(8)));

__device__ __forceinline__ void async_wait_le8() {
#if __has_builtin(__builtin_amdgcn_s_wait_asynccnt)
    __builtin_amdgcn_s_wait_asynccnt(8);
#else
    asm volatile("s_wait_asynccnt 0x8" ::: "memory");
#endif
}
__device__ __forceinline__ void async_wait_zero() {
#if __has_builtin(__builtin_amdgcn_s_wait_asynccnt)
    __builtin_amdgcn_s_wait_asynccnt(0);
#else
    asm volatile("s_wait_asynccnt 0x0" ::: "memory");
#endif
}

// One wave instruction: 32 lanes x 16B = 512B global -> 512B LDS (DMA, ASYNCcnt)
__device__ __forceinline__ void async_load_b128(unsigned lds_byte_off,
                                                const float* gaddr) {
    asm volatile("global_load_async_to_lds_b128 %0, %1, off"
                 :: "v"(lds_byte_off), "v"(gaddr)
                 : "memory");
}

// ---------------------------------------------------------------------------
// Kernel A: per (b,n): yc from det_bboxes, centers output, and row-mean of
// kernel_feats at the gathered row  -> kfm[b,n,c]
// ---------------------------------------------------------------------------
__global__ __launch_bounds__(64)
void ka_gather_mean(const float* __restrict__ kernel_feats,
                    const float* __restrict__ det_bboxes,
                    const int*   __restrict__ pad_w,
                    float* __restrict__ kfm,
                    float* __restrict__ centers)
{
    const int bn = blockIdx.x;          // 0 .. B_*N_-1
    const int b  = bn / N_;
    const float strideV = (float)pad_w[0] / (float)W_;
    const float* bb = det_bboxes + (size_t)bn * 4;
    const float yc  = (bb[1] + bb[3]) / (2.0f * strideV);
    const int   row = (int)yc;
    if (threadIdx.x == 0) centers[bn] = yc * strideV;

    const int c = threadIdx.x;          // 0..63 == channel
    const float* src = kernel_feats + (((size_t)b * C_ + c) * H_ + row) * WD_;
    float s = 0.0f;
    #pragma unroll 4
    for (int w = 0; w < WD_; ++w) s += src[w];
    kfm[(size_t)bn * C_ + c] = s * (1.0f / (float)WD_);
}

// ---------------------------------------------------------------------------
// Kernel B: per (b,n): vk = Wk*kfm + bk (65), then
//   Weff[c'] = sum_c vk[c]*Wf[c,c'] ,  beff = vk[64] + sum_c vk[c]*bf[c]
// ---------------------------------------------------------------------------
__global__ __launch_bounds__(128)
void kb_build(const float* __restrict__ kfm,
              const float* __restrict__ Wk,
              const float* __restrict__ bk,
              const float* __restrict__ Wf,
              const float* __restrict__ bf,
              float* __restrict__ Weff,
              float* __restrict__ beff)
{
    __shared__ float vk[C_ + 1];
    __shared__ float kf[C_];
    const int bn = blockIdx.x;
    const int t  = threadIdx.x;

    if (t < C_) kf[t] = kfm[(size_t)bn * C_ + t];
    __syncthreads();

    if (t < C_ + 1) {
        float s = bk[t];
        for (int c = 0; c < C_; ++c) s += Wk[t * C_ + c] * kf[c];
        vk[t] = s;
    }
    __syncthreads();

    if (t < C_) {
        float s = 0.0f;
        for (int c = 0; c < C_; ++c) s += vk[c] * Wf[c * C_ + t];
        Weff[(size_t)bn * C_ + t] = s;
    } else if (t == C_) {
        float s = vk[C_];
        for (int c = 0; c < C_; ++c) s += vk[c] * bf[c];
        beff[bn] = s;
    }
}

// ---------------------------------------------------------------------------
// Kernel C: batched 48x64 @ 64x64000 GEMM, fused bias + mask.
// V_WMMA_F32_16X16X4_F32 core; feats tiles streamed with double-buffered
// GLOBAL_LOAD_ASYNC_TO_LDS_B128 (ASYNCcnt), overlapping DMA with compute.
// ---------------------------------------------------------------------------
__global__ __launch_bounds__(256)
void kc_gemm(const float* __restrict__ feats,
             const float* __restrict__ Weff,
             const float* __restrict__ beff,
             const int*   __restrict__ image_shape,
             const int*   __restrict__ pad_w,
             float* __restrict__ logits)
{
    __shared__ float lds_f[2][C_ * LDSROW];   // 2 x 34816 B feats tiles [ch][col]
    __shared__ float lds_b[N_];               // beff

    const int b    = blockIdx.y;
    const int t    = threadIdx.x;
    const int lane = t & 31;
    const int wv   = t >> 5;              // wave 0..7
    const int mLane   = lane & 15;
    const int halfSel = lane >> 4;        // 0 | 1
    const int kHalf   = halfSel << 1;     // 0 | 2  (A/B frag K sub-offset)

    if (t < N_) lds_b[t] = beff[b * N_ + t];

    const float strideV = (float)pad_w[0] / (float)W_;
    const int hlim = (int)((float)image_shape[b * 2 + 1] / strideV);
    const int wlim = (int)((float)image_shape[b * 2 + 0] / strideV);

    // A fragments (16x4 fp32 layout: VGPR0 holds K0|K2 across lane halves,
    // VGPR1 holds K1|K3) loaded once, straight from global (L2-resident).
    const float* wsrc = Weff + (size_t)b * N_ * C_;
    v2f afrag[3][16];
    #pragma unroll
    for (int mt = 0; mt < 3; ++mt) {
        #pragma unroll
        for (int kk = 0; kk < 16; ++kk) {
            const float* p = wsrc + (mt * 16 + mLane) * C_ + kk * 4 + kHalf;
            afrag[mt][kk].x = p[0];
            afrag[mt][kk].y = p[1];
        }
    }

    const int col = wv * 16 + mLane;                 // column within tile
    const float* fb = feats  + (size_t)b * C_ * HW_;
    float*       ob = logits + (size_t)b * N_ * HW_;
    const int hwBase = blockIdx.x * (TILES_PER_BLOCK * TILE_COLS);

    // Issue one 64ch x 128col tile: 8 async B128 per wave; each instruction
    // moves one contiguous 512B channel-row chunk (ch = 8*i + wv, cols=lane*4).
    auto issue_tile = [&](int it, int bufIdx) {
        const int hw0 = hwBase + it * TILE_COLS;
        #pragma unroll
        for (int i = 0; i < 8; ++i) {
            const int ch = 8 * i + wv;
            const float* g = fb + (size_t)ch * HW_ + hw0 + lane * 4;
            const unsigned l =
                (unsigned)(size_t)&lds_f[bufIdx][ch * LDSROW + lane * 4];
            async_load_b128(l, g);
        }
    };

    issue_tile(0, 0);

    for (int it = 0; it < TILES_PER_BLOCK; ++it) {
        if (it + 1 < TILES_PER_BLOCK) {
            issue_tile(it + 1, (it + 1) & 1);   // prefetch next tile (DMA)
            async_wait_le8();                   // in-order => tile `it` landed
        } else {
            async_wait_zero();
        }
        __syncthreads();                        // tile `it` visible to all waves

        const float* buf = lds_f[it & 1];

        v8f acc0 = {0,0,0,0,0,0,0,0};
        v8f acc1 = {0,0,0,0,0,0,0,0};
        v8f acc2 = {0,0,0,0,0,0,0,0};

        #pragma unroll
        for (int kk = 0; kk < 16; ++kk) {
            v2f bfrag;
            const int r0 = (kk * 4 + kHalf) * LDSROW + col;
            bfrag.x = buf[r0];                  // K = 4kk+kHalf
            bfrag.y = buf[r0 + LDSROW];         // K = 4kk+kHalf+1
            acc0 = __builtin_amdgcn_wmma_f32_16x16x4_f32(
                       false, afrag[0][kk], false, bfrag, (short)0, acc0, false, false);
            acc1 = __builtin_amdgcn_wmma_f32_16x16x4_f32(
                       false, afrag[1][kk], false, bfrag, (short)0, acc1, false, false);
            acc2 = __builtin_amdgcn_wmma_f32_16x16x4_f32(
                       false, afrag[2][kk], false, bfrag, (short)0, acc2, false, false);
        }
        __syncthreads();   // all waves done reading buf before DMA reuses it

        // epilogue: bias + mask + store (C/D layout: VGPR r -> M = r + 8*half)
        const int hw = hwBase + it * TILE_COLS + col;
        const int h  = hw / W_;
        const int w  = hw % W_;
        const bool ok = (h < hlim) && (w < wlim);
        #pragma unroll
        for (int r = 0; r < 8; ++r) {
            const int m0 = r + 8 * halfSel;
            const float v0 = ok ? (acc0[r] + lds_b[m0])      : -1.0e8f;
            const float v1 = ok ? (acc1[r] + lds_b[m0 + 16]) : -1.0e8f;
            const float v2 = ok ? (acc2[r] + lds_b[m0 + 32]) : -1.0e8f;
            ob[(size_t)(m0)      * HW_ + hw] = v0;
            ob[(size_t)(m0 + 16) * HW_ + hw] = v1;
            ob[(size_t)(m0 + 32) * HW_ + hw] = v2;
        }
    }
}

// ---------------------------------------------------------------------------
extern "C" void kernel_launch(void* const* d_in, const int* in_sizes, int n_in,
                              void* d_out, int out_size, void* d_ws, size_t ws_size,
                              hipStream_t stream) {
    (void)in_sizes; (void)n_in; (void)out_size; (void)ws_size;

    const float* feats        = (const float*)d_in[0];
    const float* kernel_feats = (const float*)d_in[1];
    const float* Wk           = (const float*)d_in[2];
    const float* bk           = (const float*)d_in[3];
    const float* Wf           = (const float*)d_in[4];
    const float* bf           = (const float*)d_in[5];
    const float* det_bboxes   = (const float*)d_in[6];
    const int*   image_shape  = (const int*)d_in[7];
    const int*   pad_w        = (const int*)d_in[8];

    float* out     = (float*)d_out;
    float* logits  = out;                                  // B*N*H*W
    float* centers = out + (size_t)B_ * N_ * HW_;          // B*N*1

    float* ws   = (float*)d_ws;
    float* kfm  = ws;                                      // B*N*C
    float* Weff = kfm  + (size_t)B_ * N_ * C_;             // B*N*C
    float* beff = Weff + (size_t)B_ * N_ * C_;             // B*N

    ka_gather_mean<<<B_ * N_, 64, 0, stream>>>(kernel_feats, det_bboxes, pad_w,
                                               kfm, centers);
    kb_build<<<B_ * N_, 128, 0, stream>>>(kfm, Wk, bk, Wf, bf, Weff, beff);

    dim3 grid(HW_ / (TILE_COLS * TILES_PER_BLOCK), B_);    // (50, 8)
    kc_gemm<<<grid, 256, 0, stream>>>(feats, Weff, beff, image_shape, pad_w,
                                      logits);
}